// NeuralRenderer_pytorch_19138374271070
// MI455X (gfx1250) — compile-verified
//
#include <hip/hip_runtime.h>
#include <hip/hip_bf16.h>

typedef __attribute__((ext_vector_type(2))) float v2f;
typedef __attribute__((ext_vector_type(8))) float v8f;

constexpr int   kB    = 2;
constexpr int   kV    = 642;
constexpr int   kF    = 1280;
constexpr int   kF2   = 2560;   // faces + reversed-winding faces
constexpr int   kTS   = 6;
constexpr int   kHS   = 512;
constexpr int   kIMG  = 256;
constexpr float kInvFar  = 0.01f;   // 1/FAR
constexpr float kInvNear = 10.0f;   // 1/NEAR
constexpr float kVW      = 0.57735027f; // tan(30 deg)

// ---------------- K0: vertex transform (quat rotate + camera + projection) ----
__global__ void vert_kernel(const float* __restrict__ verts,
                            const float* __restrict__ cams,
                            float* __restrict__ vpB,   // view-space pos (for lighting)
                            float* __restrict__ pjB) { // projected (X, Y, Zc)
  int t = blockIdx.x * blockDim.x + threadIdx.x;
  if (t >= kB * kV) return;
  int b = t / kV;
  const float* X = verts + (size_t)t * 3;
  const float* c = cams + (size_t)b * 7;
  float s = c[0], trx = c[1], trY = c[2];
  float qw = c[3], qx = c[4], qy = c[5], qz = c[6];
  float x = X[0], y = X[1], z = X[2];
  // t = 2*cross(u, X); Xr = X + w*t + cross(u, t)
  float tX = 2.0f * (qy * z - qz * y);
  float tY = 2.0f * (qz * x - qx * z);
  float tZ = 2.0f * (qx * y - qy * x);
  float rx = x + qw * tX + (qy * tZ - qz * tY);
  float ry = y + qw * tY + (qz * tX - qx * tZ);
  float rz = z + qw * tZ + (qx * tY - qy * tX);
  float vpx = s * rx + trx;
  float vpy = -(s * ry + trY);          // y-flip applied in reference
  float vpz = s * rz + 5.0f;            // OFFSET_Z
  vpB[t * 3 + 0] = vpx; vpB[t * 3 + 1] = vpy; vpB[t * 3 + 2] = vpz;
  // R_CAM == identity, EYE = (0,0,-2.732) -> vc = vp - EYE
  float zc = vpz + 2.732f;
  float d = zc * kVW;
  pjB[t * 3 + 0] = vpx / d;
  pjB[t * 3 + 1] = vpy / d;
  pjB[t * 3 + 2] = zc;
}

// ---------------- K1: per-face edge-function coefficients + lighting ---------
// Record (16 floats, 64B): [A0,B0, A1,B1, A2,B2, An,Bn,  C0,C1,C2,Cn,  light, 0,0,0]
__global__ void face_kernel(const int* __restrict__ faces,
                            const float* __restrict__ vpB,
                            const float* __restrict__ pjB,
                            float* __restrict__ rec) {
  int t = blockIdx.x * blockDim.x + threadIdx.x;
  if (t >= kB * kF2) return;
  int b = t / kF2;
  int f2 = t - b * kF2;
  int i0, i1, i2;
  if (f2 < kF) {
    const int* fp = faces + ((size_t)b * kF + f2) * 3;
    i0 = fp[0]; i1 = fp[1]; i2 = fp[2];
  } else {                                   // reversed winding half
    const int* fp = faces + ((size_t)b * kF + (f2 - kF)) * 3;
    i0 = fp[2]; i1 = fp[1]; i2 = fp[0];
  }
  const float* v0 = vpB + ((size_t)b * kV + i0) * 3;
  const float* v1 = vpB + ((size_t)b * kV + i1) * 3;
  const float* v2 = vpB + ((size_t)b * kV + i2) * 3;
  float ax = v2[0] - v1[0], ay = v2[1] - v1[1], az = v2[2] - v1[2];
  float bx = v0[0] - v1[0], by = v0[1] - v1[1], bz = v0[2] - v1[2];
  float nx = ay * bz - az * by;
  float ny = az * bx - ax * bz;
  float nz = ax * by - ay * bx;
  float nrm = fmaxf(sqrtf(nx * nx + ny * ny + nz * nz), 1e-5f);
  float light = 0.5f + 0.5f * fmaxf(ny / nrm, 0.0f);

  const float* p0 = pjB + ((size_t)b * kV + i0) * 3;
  const float* p1 = pjB + ((size_t)b * kV + i1) * 3;
  const float* p2 = pjB + ((size_t)b * kV + i2) * 3;
  float x0p = p0[0], y0p = p0[1], z0p = p0[2];
  float x1p = p1[0], y1p = p1[1], z1p = p1[2];
  float x2p = p2[0], y2p = p2[1], z2p = p2[2];
  // e(i,j)(px,py) = (yi-yj)*px + (xj-xi)*py + (xi*yj - xj*yi)
  float A0 = y1p - y2p, B0 = x2p - x1p, C0 = x1p * y2p - x2p * y1p; // e(1,2)
  float A1 = y2p - y0p, B1 = x0p - x2p, C1 = x2p * y0p - x0p * y2p; // e(2,0)
  float A2 = y0p - y1p, B2 = x1p - x0p, C2 = x0p * y1p - x1p * y0p; // e(0,1)
  float rz0 = 1.0f / z0p, rz1 = 1.0f / z1p, rz2 = 1.0f / z2p;
  float An = A0 * rz0 + A1 * rz1 + A2 * rz2;
  float Bn = B0 * rz0 + B1 * rz1 + B2 * rz2;
  float Cn = C0 * rz0 + C1 * rz1 + C2 * rz2;
  float* r = rec + (size_t)t * 16;
  r[0] = A0; r[1] = B0; r[2] = A1; r[3] = B1;
  r[4] = A2; r[5] = B2; r[6] = An; r[7] = Bn;
  r[8] = C0; r[9] = C1; r[10] = C2; r[11] = Cn;
  r[12] = light; r[13] = 0.0f; r[14] = 0.0f; r[15] = 0.0f;
}

// ---------------- K2: WMMA rasterizer (z-buffer via max-invz) ----------------
// One wave owns 16 consecutive pixels in one row; loops over all 2560 faces in
// blocks of 16. Per block: 4x V_WMMA_F32_16X16X4_F32 evaluate e0,e1,e2,num for
// 16 faces x 16 pixels. D layout: lane n%16 = pixel, vgpr m + 8*(lane/16) = face.
// Inner loop is fully branchless (v_cmp/v_cndmask/min3/max3 only).
__global__ __launch_bounds__(256) void raster_kernel(const float* __restrict__ rec,
                                                     int* __restrict__ fidx) {
  const int lane = threadIdx.x & 31;
  const int wave = blockIdx.x * 8 + (threadIdx.x >> 5);
  const int b    = wave / (kHS * (kHS / 16));
  const int rem  = wave - b * (kHS * (kHS / 16));
  const int row  = rem >> 5;          // 32 pixel-groups per row
  const int x0   = (rem & 31) << 4;
  const bool lo  = lane < 16;
  const int  n   = lane & 15;

  const float px = (2.0f * (float)(x0 + n) + 1.0f - (float)kHS) * (1.0f / (float)kHS);
  const float py = (2.0f * (float)row + 1.0f - (float)kHS) * (1.0f / (float)kHS);
  // B matrix 4x16: rows (px, py, 1, 0); lanes 0-15 hold K=0,1; lanes 16-31 K=2,3
  v2f bv;
  bv.x = lo ? px : 1.0f;
  bv.y = lo ? py : 0.0f;

  const float* recBase = rec + (size_t)b * kF2 * 16;
  float best = 0.0f;     // max clamped invz among ok faces (== min depth)
  int bestIdx = -1;

  for (int blk = 0; blk < kF2 / 16; ++blk) {
    const float4* rp = (const float4*)(recBase + (size_t)(blk * 16 + n) * 16);
    float4 r0 = rp[0], r1 = rp[1], r2 = rp[2];
    int nb = (blk + 1 < kF2 / 16) ? blk + 1 : blk;
    __builtin_prefetch(recBase + (size_t)(nb * 16 + n) * 16, 0, 1);

    // A matrix 16x4 per edge function: lanes 0-15 hold K=0,1 (A,B),
    // lanes 16-31 hold K=2,3 (C, 0)
    v2f a0, a1, a2, an;
    a0.x = lo ? r0.x : r2.x;  a0.y = lo ? r0.y : 0.0f;
    a1.x = lo ? r0.z : r2.y;  a1.y = lo ? r0.w : 0.0f;
    a2.x = lo ? r1.x : r2.z;  a2.y = lo ? r1.y : 0.0f;
    an.x = lo ? r1.z : r2.w;  an.y = lo ? r1.w : 0.0f;

    v8f zero = {};
    v8f E0 = __builtin_amdgcn_wmma_f32_16x16x4_f32(false, a0, false, bv, (short)0, zero, false, false);
    v8f E1 = __builtin_amdgcn_wmma_f32_16x16x4_f32(false, a1, false, bv, (short)0, zero, false, false);
    v8f E2 = __builtin_amdgcn_wmma_f32_16x16x4_f32(false, a2, false, bv, (short)0, zero, false, false);
    v8f NU = __builtin_amdgcn_wmma_f32_16x16x4_f32(false, an, false, bv, (short)0, zero, false, false);

    const int faceBase = blk * 16 + (lo ? 0 : 8);
#pragma unroll
    for (int m = 0; m < 8; ++m) {
      float e0 = E0[m], e1 = E1[m], e2 = E2[m], nu = NU[m];
      float den = e0 + e1 + e2;
      bool denok = fabsf(den) > 1e-10f;
      float ds = denok ? den : 1e-10f;
      // inside-test, branch-free: all w_i = e_i/den >= 0  <=>  all e_i share
      // den's sign <=> min3 >= 0 (=> den >= 0) or max3 <= 0 (=> den <= 0);
      // den == 0 case is excluded by denok.
      float mn = fminf(e0, fminf(e1, e2));
      float mx = fmaxf(e0, fmaxf(e1, e2));
      bool geom = (mn >= 0.0f) | (mx <= 0.0f);
      float invz = nu / ds;
      float invzc = (fabsf(invz) > 1e-10f) ? invz : 1e-10f;
      // NEAR < zp < FAR  <=>  1/FAR < invzc < 1/NEAR ; min-depth <=> max-invz
      bool ok = denok & geom & (invzc > kInvFar) & (invzc < kInvNear);
      bool upd = ok & (invzc > best);
      best    = upd ? invzc : best;
      bestIdx = upd ? (faceBase + m) : bestIdx;
    }
  }
  // merge lane-halves (upper half holds faces +8 of each block); preserve
  // lowest-face-index tie-break of the reference scan (branch-free)
  float obest = __shfl_xor(best, 16, 32);
  int   oidx  = __shfl_xor(bestIdx, 16, 32);
  bool take = (obest > best) |
              ((obest == best) & (oidx >= 0) & ((bestIdx < 0) | (oidx < bestIdx)));
  bestIdx = take ? oidx : bestIdx;
  if (lo) fidx[((size_t)b * kHS + row) * kHS + (x0 + n)] = bestIdx;
}

// ---------------- K3: barycentric texture shade + 2x2 downsample -------------
__global__ __launch_bounds__(256) void shade_kernel(const float* __restrict__ rec,
                                                    const int* __restrict__ fidx,
                                                    const float* __restrict__ tex,
                                                    float* __restrict__ out) {
  int t = blockIdx.x * blockDim.x + threadIdx.x;   // kB*kIMG*kIMG threads
  int b = t / (kIMG * kIMG);
  int rem = t - b * (kIMG * kIMG);
  int i = rem >> 8;
  int j = rem & 255;
  float acc0 = 0.0f, acc1 = 0.0f, acc2 = 0.0f;
#pragma unroll
  for (int s = 0; s < 4; ++s) {
    int p = 2 * i + (s >> 1);
    int q = 2 * j + (s & 1);
    int f2 = fidx[((size_t)b * kHS + p) * kHS + q];
    if (f2 < 0) continue;
    const float* r = rec + ((size_t)b * kF2 + f2) * 16;
    float px = (2.0f * (float)q + 1.0f - (float)kHS) * (1.0f / (float)kHS);
    float py = (2.0f * (float)p + 1.0f - (float)kHS) * (1.0f / (float)kHS);
    float e0 = fmaf(r[0], px, fmaf(r[1], py, r[8]));
    float e1 = fmaf(r[2], px, fmaf(r[3], py, r[9]));
    float e2 = fmaf(r[4], px, fmaf(r[5], py, r[10]));
    float den = e0 + e1 + e2;
    float ds = (fabsf(den) > 1e-10f) ? den : 1e-10f;
    float w0 = e0 / ds, w1 = e1 / ds, w2 = e2 / ds;
    int t0 = (int)fminf(fmaxf(floorf(w0 * (float)kTS), 0.0f), (float)(kTS - 1));
    int t1 = (int)fminf(fmaxf(floorf(w1 * (float)kTS), 0.0f), (float)(kTS - 1));
    int t2 = (int)fminf(fmaxf(floorf(w2 * (float)kTS), 0.0f), (float)(kTS - 1));
    float light = r[12];
    int fb, d0, d1, d2;
    if (f2 < kF) { fb = f2;      d0 = t0; d1 = t1; d2 = t2; }
    else         { fb = f2 - kF; d0 = t2; d1 = t1; d2 = t0; } // transposed tex half
    const float* tp = tex + (((size_t)b * kF + fb) * (kTS * kTS * kTS)
                             + (d0 * kTS * kTS + d1 * kTS + d2)) * 3;
    acc0 += tp[0] * light;
    acc1 += tp[1] * light;
    acc2 += tp[2] * light;
  }
  size_t o = ((size_t)b * 3) * (kIMG * kIMG) + (size_t)i * kIMG + j;
  out[o]                     = acc0 * 0.25f;
  out[o + kIMG * kIMG]       = acc1 * 0.25f;
  out[o + 2 * kIMG * kIMG]   = acc2 * 0.25f;
}

extern "C" void kernel_launch(void* const* d_in, const int* in_sizes, int n_in,
                              void* d_out, int out_size, void* d_ws, size_t ws_size,
                              hipStream_t stream) {
  (void)in_sizes; (void)n_in; (void)out_size; (void)ws_size;
  const float* verts = (const float*)d_in[0];
  const int*   faces = (const int*)d_in[1];
  const float* cams  = (const float*)d_in[2];
  const float* tex   = (const float*)d_in[3];
  float* out = (float*)d_out;

  // workspace layout (floats): vp[2*642*3] | pj[2*642*3] | rec[2*2560*16] | fidx[2*512*512]
  float* vpB = (float*)d_ws;
  float* pjB = vpB + (size_t)kB * kV * 3;
  float* rec = pjB + (size_t)kB * kV * 3;       // 32B-aligned (offset 30816 B)
  int*   fid = (int*)(rec + (size_t)kB * kF2 * 16);

  vert_kernel<<<(kB * kV + 255) / 256, 256, 0, stream>>>(verts, cams, vpB, pjB);
  face_kernel<<<(kB * kF2 + 255) / 256, 256, 0, stream>>>(faces, vpB, pjB, rec);
  raster_kernel<<<(kB * kHS * (kHS / 16)) / 8, 256, 0, stream>>>(rec, fid);
  shade_kernel<<<(kB * kIMG * kIMG) / 256, 256, 0, stream>>>(rec, fid, tex, out);
}